// VectorQuantizer_18769007083533
// MI455X (gfx1250) — compile-verified
//
#include <hip/hip_runtime.h>
#include <hip/hip_bf16.h>

typedef __attribute__((ext_vector_type(2))) float v2f;
typedef __attribute__((ext_vector_type(8))) float v8f;

#define NUM_EMB   1024
#define EMB_DIM   256
#define HW        1024          // 32*32 spatial
#define N_ROWS    32768         // 32*32*32
#define E_PITCH   260           // LDS pitch: breaks stride-256 bank conflicts

// d_out layout (floats), concatenated in reference return order
#define O_LOSS 0
#define O_ZQ   1
#define O_ENC  (1 + 32 * 256 * 1024)                 // after z_q (8388608)
#define O_IDX  (O_ENC + 32768 * 1024)                // after min_encodings (33554432)

// ---------------------------------------------------------------- kernel 0
// Precompute ||e_k||^2 for all 1024 codebook rows into workspace.
__global__ void __launch_bounds__(256)
vq_enorm_kernel(const float* __restrict__ emb, float* __restrict__ enorm) {
    int k = blockIdx.x * 256 + threadIdx.x;
    if (k < NUM_EMB) {
        const float* e = emb + (size_t)k * EMB_DIM;
        float s = 0.f;
        #pragma unroll 8
        for (int d = 0; d < EMB_DIM; ++d) s += e[d] * e[d];
        enorm[k] = s;
    }
}

// ---------------------------------------------------------------- kernel 1
// Each wave owns a 16-row M-tile of z_flat (A panel fully register-resident,
// 128 VGPRs/lane). Block of 8 waves shares LDS-staged 16-wide codebook tiles.
// Distance GEMM runs on V_WMMA_F32_16X16X4_F32 (fp32 matrix core path).
__global__ void __launch_bounds__(256)
vq_main_kernel(const float* __restrict__ z, const float* __restrict__ emb,
               const float* __restrict__ enorm, float* __restrict__ out,
               float* __restrict__ blockPartial)
{
    __shared__ float eTile[16 * E_PITCH];   // 16 codebook rows, padded pitch
    __shared__ float enLds[NUM_EMB];        // ||e||^2 cache
    __shared__ float wavePart[8];

    const int tid  = threadIdx.x;
    const int wave = tid >> 5;
    const int lane = tid & 31;
    const int l16  = lane & 15;
    const int half = lane >> 4;

    // stage ||e||^2 into LDS
    for (int i = tid; i < NUM_EMB; i += 256) enLds[i] = enorm[i];

    const int tile = blockIdx.x * 8 + wave;   // 0..2047, 16 rows each
    const int n0   = tile * 16;
    const int b0   = n0 >> 10;                // batch index (16 | 1024 => no split)
    const int p0   = n0 & 1023;               // spatial index base

    // ---- load A panel: z_flat[n0+l16, :] per WMMA 16x16x4 fp32 A layout:
    // lanes 0-15 hold (K=0,K=1) of each d-step, lanes 16-31 hold (K=2,K=3).
    const float* zp = z + (size_t)b0 * (EMB_DIM * HW) + p0 + l16;
    v2f a_frag[64];
    #pragma unroll
    for (int s = 0; s < 64; ++s) {
        const int d = 4 * s + 2 * half;
        a_frag[s].x = zp[(size_t)d       * HW];
        a_frag[s].y = zp[(size_t)(d + 1) * HW];
    }

    float best[8];
    int   bestk[8];
    #pragma unroll
    for (int r = 0; r < 8; ++r) { best[r] = 3.4e38f; bestk[r] = 0; }

    // B fragment LDS base: lanes 0-15 read (K=0,1), lanes 16-31 read (K=2,3)
    const int bOff = l16 * E_PITCH + 2 * half;

    for (int kt = 0; kt < 64; ++kt) {
        const int k0 = kt * 16;
        __syncthreads();
        // cooperative stage of E[k0 .. k0+16), 16KB, coalesced
        for (int i = tid; i < 16 * EMB_DIM; i += 256) {
            const int kk = i >> 8, d = i & 255;
            eTile[kk * E_PITCH + d] = emb[(size_t)(k0 + kk) * EMB_DIM + d];
        }
        __syncthreads();

        v8f c0 = {0.f,0.f,0.f,0.f,0.f,0.f,0.f,0.f};
        v8f c1 = {0.f,0.f,0.f,0.f,0.f,0.f,0.f,0.f};
        #pragma unroll
        for (int s = 0; s < 64; s += 2) {
            v2f bA = *(const v2f*)(&eTile[bOff + 4 * s]);
            v2f bB = *(const v2f*)(&eTile[bOff + 4 * (s + 1)]);
            c0 = __builtin_amdgcn_wmma_f32_16x16x4_f32(
                    false, a_frag[s],     false, bA, (short)0, c0, false, false);
            c1 = __builtin_amdgcn_wmma_f32_16x16x4_f32(
                    false, a_frag[s + 1], false, bB, (short)0, c1, false, false);
        }

        // score = ||e||^2 - 2*dot   (||z||^2 constant across k, drop it)
        const float en = enLds[k0 + l16];
        #pragma unroll
        for (int r = 0; r < 8; ++r) {
            const float score = en - 2.0f * (c0[r] + c1[r]);
            if (score < best[r]) { best[r] = score; bestk[r] = k0 + l16; }
        }
    }

    // butterfly min-with-index within each 16-lane half (prefer smaller k on tie)
    #pragma unroll
    for (int r = 0; r < 8; ++r) {
        #pragma unroll
        for (int m = 8; m >= 1; m >>= 1) {
            const float os = __shfl_xor(best[r],  m, 32);
            const int   ok = __shfl_xor(bestk[r], m, 32);
            if (os < best[r] || (os == best[r] && ok < bestk[r])) {
                best[r] = os; bestk[r] = ok;
            }
        }
    }

    // ------------------------------------------------ epilogue: 16 rows/wave
    float lsum = 0.f;
    for (int row = 0; row < 16; ++row) {
        const int idx = __shfl(bestk[row & 7], (row < 8) ? 0 : 16, 32);
        const int n = n0 + row;
        const int p = p0 + row;

        if (lane == 0) out[O_IDX + n] = (float)idx;

        // one-hot row: pure streaming store, keep out of L2
        float* encRow = out + O_ENC + (size_t)n * NUM_EMB;
        #pragma unroll
        for (int j = lane; j < NUM_EMB; j += 32)
            __builtin_nontemporal_store((j == idx) ? 1.0f : 0.0f, encRow + j);

        // z_q gather (NCHW) + commitment/codebook loss contribution
        const float* er  = emb + (size_t)idx * EMB_DIM;
        const float* zr  = z   + (size_t)b0 * (EMB_DIM * HW) + p;
        float*       zqr = out + O_ZQ + (size_t)b0 * (EMB_DIM * HW) + p;
        #pragma unroll
        for (int d = lane; d < EMB_DIM; d += 32) {
            const float e  = er[d];
            const float zv = zr[(size_t)d * HW];
            __builtin_nontemporal_store(e, zqr + (size_t)d * HW);
            const float df = e - zv;
            lsum += df * df;
        }
    }

    // deterministic block partial: wave butterfly -> LDS -> ordered sum
    #pragma unroll
    for (int m = 16; m >= 1; m >>= 1) lsum += __shfl_xor(lsum, m, 32);
    if (lane == 0) wavePart[wave] = lsum;
    __syncthreads();
    if (tid == 0) {
        float s = 0.f;
        #pragma unroll
        for (int w = 0; w < 8; ++w) s += wavePart[w];
        blockPartial[blockIdx.x] = s;
    }
}

// ---------------------------------------------------------------- kernel 2
// Deterministic final reduction of 256 block partials -> loss scalar.
__global__ void vq_loss_finalize(const float* __restrict__ part,
                                 float* __restrict__ out) {
    if (threadIdx.x == 0 && blockIdx.x == 0) {
        double s = 0.0;
        for (int i = 0; i < 256; ++i) s += (double)part[i];
        // loss = mean((zq-z)^2) + BETA*mean((zq-z)^2) = 1.25 * sum / numel
        out[O_LOSS] = (float)(1.25 * s / 8388608.0);
    }
}

extern "C" void kernel_launch(void* const* d_in, const int* in_sizes, int n_in,
                              void* d_out, int out_size, void* d_ws, size_t ws_size,
                              hipStream_t stream) {
    const float* z   = (const float*)d_in[0];   // [32,256,32,32] fp32
    const float* emb = (const float*)d_in[1];   // [1024,256] fp32
    float* out = (float*)d_out;

    float* enorm    = (float*)d_ws;             // 1024 floats
    float* partials = enorm + NUM_EMB;          // 256 floats

    vq_enorm_kernel<<<4, 256, 0, stream>>>(emb, enorm);
    vq_main_kernel<<<256, 256, 0, stream>>>(z, emb, enorm, out, partials);
    vq_loss_finalize<<<1, 32, 0, stream>>>(partials, out);
}